// QuantumSuperpositionLayer_37649683316879
// MI455X (gfx1250) — compile-verified
//
#include <hip/hip_runtime.h>
#include <math.h>

typedef __attribute__((ext_vector_type(2))) float v2f;
typedef __attribute__((ext_vector_type(8))) float v8f;
typedef __attribute__((ext_vector_type(4))) unsigned int v4u;
typedef __attribute__((ext_vector_type(4))) int v4i;
typedef __attribute__((ext_vector_type(8))) int v8i;

#define NQ      16
#define KDIM    4096     /* S / NQ */
#define D       128
#define BATCH   4
#define ENTW    0.8f
#define NBLOCKS 128
#define KCHUNK  32       /* KDIM / NBLOCKS */
#define KTD     4        /* k-rows per TDM tile */
#define NCHUNK  (KCHUNK / KTD)

/* workspace layout (units of 4 bytes) */
#define OFF_EWP    0                  /* 256 f32 : coh*(ENT*softmax(E)+(1-ENT)*I), [q][p] */
#define OFF_PHASE  256                /* 2 f32   : cos(phi), sin(phi)                    */
#define OFF_BIAS   272                /* 128 f32 : sum_n meas_b[n][e]                    */
#define OFF_PSCORE 512                /* 128*512 f32 block-partial best scores           */
#define OFF_PIDX   (512 + 65536)      /* 128*512 u32 block-partial best indices          */
#define OFF_IDX    (512 + 131072)     /* 512 u32 final argmax indices [b][d]             */

/* LDS: 8 waves * 4096 floats (2 bufs * (1024 re + 1024 im)) + 1024-entry gumbel table */
#define LDS_WAVE_F   4096
#define LDS_BUF_F    2048
#define LDS_IM_F     1024
#define LDS_EGT_F    32768
#define LDS_TOTAL_F  33792

__device__ __forceinline__ unsigned int mix32(unsigned int h) {
  h ^= h >> 16; h *= 0x7FEB352Du;
  h ^= h >> 15; h *= 0x846CA68Bu;
  h ^= h >> 16;
  return h;
}

/* ---- TDM: load a (dim0=16 d, dim1=KTD k, dim2=16 p) f32 tile into LDS ---- */
__device__ __forceinline__ void tdm_load_tile(unsigned lds_addr, const float* gptr) {
  unsigned long long ga = (unsigned long long)(uintptr_t)gptr;
  v4u g0;
  g0[0] = 1u;                                   /* count=1, user mode           */
  g0[1] = lds_addr;                             /* LDS byte address             */
  g0[2] = (unsigned)ga;                         /* global_addr[31:0]            */
  g0[3] = (unsigned)(ga >> 32) | (2u << 30);    /* global_addr[56:32] | type=2  */
  v8i g1;
  g1[0] = 0x20000;                              /* data_size=4B, wg_mask=0      */
  g1[1] = (int)(16u << 16);                     /* tensor_dim0 = 16             */
  g1[2] = (int)((unsigned)KTD << 16);           /* tensor_dim1 = KTD            */
  g1[3] = (int)(16u << 16);                     /* tile_dim0 = 16               */
  g1[4] = (int)((unsigned)KTD | (16u << 16));   /* tile_dim1=KTD, tile_dim2=16  */
  g1[5] = 128;                                  /* tensor_dim0_stride = 128     */
  g1[6] = 0;                                    /* stride0 hi / stride1 lo      */
  g1[7] = 8;                                    /* stride1[47:16]: 524288>>16   */
  v4i g2;
  g2[0] = 16;                                   /* tensor_dim2 = 16             */
  g2[1] = 0; g2[2] = 0; g2[3] = 0;
  v4i g3 = {0, 0, 0, 0};
#if __has_include(<hip/amd_detail/amd_gfx1250_TDM.h>)
  v8i g4 = {0, 0, 0, 0, 0, 0, 0, 0};            /* therock 6-arg form */
  __builtin_amdgcn_tensor_load_to_lds(g0, g1, g2, g3, g4, 0);
#else
  __builtin_amdgcn_tensor_load_to_lds(g0, g1, g2, g3, 0);  /* ROCm 7.2 5-arg */
#endif
}

/* ---- tiny prep: phase, folded mix matrix, bias sum ---- */
__global__ void qsl_prep(const float* __restrict__ ang,
                         const float* __restrict__ ent,
                         const float* __restrict__ mb,
                         float* __restrict__ ws) {
  const int t = threadIdx.x;
  const float coh = __expf(-1.0f / 100.0f);
  if (t == 0) {
    float s = 0.0f;
    for (int i = 0; i < NQ * 3; ++i) s += ang[i];
    ws[OFF_PHASE + 0] = __cosf(s);
    ws[OFF_PHASE + 1] = __sinf(s);
  }
  if (t < NQ) {
    float row[NQ];
    float m = -3.0e38f;
    for (int p = 0; p < NQ; ++p) { row[p] = ent[t * NQ + p]; m = fmaxf(m, row[p]); }
    float sum = 0.0f;
    for (int p = 0; p < NQ; ++p) { row[p] = __expf(row[p] - m); sum += row[p]; }
    const float inv = 1.0f / sum;
    for (int p = 0; p < NQ; ++p) {
      float w = ENTW * row[p] * inv + ((p == t) ? (1.0f - ENTW) : 0.0f);
      ws[OFF_EWP + t * NQ + p] = coh * w;
    }
  }
  if (t < D) {
    float a = 0.0f;
    for (int n = 0; n < NQ; ++n) a += mb[n * D + t];
    ws[OFF_BIAS + t] = a;
  }
}

/* ---- main streaming pass: TDM double-buffer -> WMMA mix -> table-Gumbel argmax ---- */
__global__ void __launch_bounds__(256) qsl_main(const float* __restrict__ qre,
                                                const float* __restrict__ qim,
                                                float* __restrict__ ws) {
  __shared__ float smem[LDS_TOTAL_F];
  const int tid   = threadIdx.x;
  const int lane  = tid & 31;
  const int wv    = __builtin_amdgcn_readfirstlane(tid >> 5);  /* wave-uniform */
  const int hl    = lane >> 4;
  const int dcol  = lane & 15;
  const int d     = wv * 16 + dcol;
  const int kbase = blockIdx.x * KCHUNK;

  /* gumbel table: eg(u) = -1/log(u) so that power*eg ~ exp(log|amp|^2 + gumbel) */
  for (int i = tid; i < 1024; i += 256)
    smem[LDS_EGT_F + i] = -1.0f / __logf((float)i * (1.0f / 1024.0f) + (0.5f / 1024.0f));

  /* A operand: ewp in WMMA A-layout (M=q across lanes%16, K split over regs/halves) */
  const int qa = lane & 15;
  v2f a[4];
#pragma unroll
  for (int c = 0; c < 4; ++c) {
    const int p0 = 4 * c + 2 * hl;
    a[c].x = ws[OFF_EWP + qa * NQ + p0];
    a[c].y = ws[OFF_EWP + qa * NQ + p0 + 1];
  }
  const float cphi = ws[OFF_PHASE + 0];
  const float sphi = ws[OFF_PHASE + 1];

  /* issue chunk 0 TDM (per-wave tile, per-wave counter) */
  const unsigned ldsbase = (unsigned)wv * (LDS_WAVE_F * 4u);
  {
    const size_t go = (size_t)kbase * D + (size_t)wv * 16;
    tdm_load_tile(ldsbase + 0u,                qre + go);
    tdm_load_tile(ldsbase + LDS_IM_F * 4u,     qim + go);
  }
  __syncthreads();   /* egtab visible to all waves */

  float bscore[BATCH];
  unsigned int bidx[BATCH];
#pragma unroll
  for (int b = 0; b < BATCH; ++b) { bscore[b] = -3.0e38f; bidx[b] = 0u; }

  const float NAMP = 0.70710678f * (1.0f - 0.99004983f) * 0.1f * 1.7320508f;
  const float S1   = NAMP * (2.0f / 255.0f);
  const unsigned UMUL[BATCH] = {0x9E3779B1u, 0x85EBCA77u, 0xC2B2AE3Du, 0x27D4EB2Fu};
  const float* eg = &smem[LDS_EGT_F];

  for (int c = 0; c < NCHUNK; ++c) {
    if (c + 1 < NCHUNK) {  /* uniform branch: prefetch next chunk via TDM */
      const size_t go = (size_t)(kbase + (c + 1) * KTD) * D + (size_t)wv * 16;
      const unsigned lb = ldsbase + (unsigned)((c + 1) & 1) * (LDS_BUF_F * 4u);
      tdm_load_tile(lb, qre + go);
      tdm_load_tile(lb + LDS_IM_F * 4u, qim + go);
      __builtin_amdgcn_s_wait_tensorcnt((short)2);  /* oldest 2 (chunk c) done */
    } else {
      __builtin_amdgcn_s_wait_tensorcnt((short)0);
    }
    const float* bre_base = &smem[wv * LDS_WAVE_F + (c & 1) * LDS_BUF_F];
    const float* bim_base = bre_base + LDS_IM_F;

    for (int kk = 0; kk < KTD; ++kk) {
      v2f br[4], bi[4];
#pragma unroll
      for (int cc = 0; cc < 4; ++cc) {           /* B operands from LDS tile */
        const int p0 = 4 * cc + 2 * hl;
        const int o0 = (p0 * KTD + kk) * 16 + dcol;
        br[cc].x = bre_base[o0];
        br[cc].y = bre_base[o0 + KTD * 16];
        bi[cc].x = bim_base[o0];
        bi[cc].y = bim_base[o0 + KTD * 16];
      }
      v8f accre = {}, accim = {};
#pragma unroll
      for (int cc = 0; cc < 4; ++cc) {
        accre = __builtin_amdgcn_wmma_f32_16x16x4_f32(false, a[cc], false, br[cc],
                                                      (short)0, accre, false, false);
        accim = __builtin_amdgcn_wmma_f32_16x16x4_f32(false, a[cc], false, bi[cc],
                                                      (short)0, accim, false, false);
      }
      const int k = kbase + c * KTD + kk;
#pragma unroll
      for (int r = 0; r < 8; ++r) {
        const float mr = accre[r] * cphi - accim[r] * sphi;
        const float mi = accre[r] * sphi + accim[r] * cphi;
        const unsigned q    = (unsigned)(r + 8 * hl);
        const unsigned sidx = q * KDIM + (unsigned)k;
        const unsigned seed = sidx * 0x9E3779B1u ^ (unsigned)d * 0x85EBCA77u;
        const unsigned h0 = mix32(seed);
        const unsigned h1 = mix32(seed ^ 0x68E31DA4u);
        const float mro = mr - NAMP;
        const float mio = mi - NAMP;
#pragma unroll
        for (int b = 0; b < BATCH; ++b) {
          const float rr = fmaf((float)((h0 >> (8 * b)) & 255u), S1, mro);
          const float ii = fmaf((float)((h1 >> (8 * b)) & 255u), S1, mio);
          const float pw = fmaf(ii, ii, rr * rr);
          const unsigned u10 = (h1 * UMUL[b]) >> 22;
          const float score = pw * eg[u10];
          const bool better = score > bscore[b];
          bscore[b] = better ? score : bscore[b];
          bidx[b]   = better ? sidx  : bidx[b];
        }
      }
    }
  }

  /* lanes L and L+16 hold the same d for the two q-halves: merge across halves */
#pragma unroll
  for (int b = 0; b < BATCH; ++b) {
    const float        os = __shfl_xor(bscore[b], 16, 32);
    const unsigned int oi = (unsigned int)__shfl_xor((int)bidx[b], 16, 32);
    const bool take = (os > bscore[b]) || (os == bscore[b] && oi < bidx[b]);
    bscore[b] = take ? os : bscore[b];
    bidx[b]   = take ? oi : bidx[b];
  }
  if (lane < 16) {
#pragma unroll
    for (int b = 0; b < BATCH; ++b) {
      const size_t o = (size_t)blockIdx.x * 512 + (size_t)b * 128 + d;
      ws[OFF_PSCORE + o] = bscore[b];
      ((unsigned int*)ws)[OFF_PIDX + o] = bidx[b];
    }
  }
}

/* ---- reduce 128 block partials -> final argmax index per (b,d) ---- */
__global__ void qsl_reduce(float* __restrict__ ws) {
  const int tid = blockIdx.x * blockDim.x + threadIdx.x;   /* 0..511 = b*128+d */
  if (tid >= BATCH * D) return;
  float best = -3.0e38f;
  unsigned int bi = 0u;
  for (int blk = 0; blk < NBLOCKS; ++blk) {
    const float        s = ws[OFF_PSCORE + (size_t)blk * 512 + tid];
    const unsigned int i = ((const unsigned int*)ws)[OFF_PIDX + (size_t)blk * 512 + tid];
    const bool take = (s > best) || (s == best && i < bi);
    best = take ? s : best;
    bi   = take ? i : bi;
  }
  ((unsigned int*)ws)[OFF_IDX + tid] = bi;
}

/* ---- measurement gather + bias mean ---- */
__global__ void qsl_out(const float* __restrict__ mW,
                        const float* __restrict__ ws,
                        float* __restrict__ out) {
  const int tid = blockIdx.x * blockDim.x + threadIdx.x;   /* 0..511 = b*128+e */
  if (tid >= BATCH * D) return;
  const int b = tid >> 7;
  const int e = tid & 127;
  float acc = ws[OFF_BIAS + e];
  const unsigned int* idx = ((const unsigned int*)ws) + OFF_IDX + b * D;
  for (int dd = 0; dd < D; ++dd) {
    const unsigned int i = idx[dd];
    if (i < NQ) acc += mW[((size_t)i * D + e) * D + dd];
  }
  out[tid] = acc * (1.0f / 16.0f);
}

extern "C" void kernel_launch(void* const* d_in, const int* in_sizes, int n_in,
                              void* d_out, int out_size, void* d_ws, size_t ws_size,
                              hipStream_t stream) {
  const float* qre = (const float*)d_in[1];
  const float* qim = (const float*)d_in[2];
  const float* ang = (const float*)d_in[3];
  const float* ent = (const float*)d_in[4];
  const float* mW  = (const float*)d_in[5];
  const float* mb  = (const float*)d_in[6];
  float* ws  = (float*)d_ws;
  float* out = (float*)d_out;

  hipLaunchKernelGGL(qsl_prep,   dim3(1),       dim3(256), 0, stream, ang, ent, mb, ws);
  hipLaunchKernelGGL(qsl_main,   dim3(NBLOCKS), dim3(256), 0, stream, qre, qim, ws);
  hipLaunchKernelGGL(qsl_reduce, dim3(2),       dim3(256), 0, stream, ws);
  hipLaunchKernelGGL(qsl_out,    dim3(2),       dim3(256), 0, stream, mW, ws, out);
}